// SAPS_72696616452335
// MI455X (gfx1250) — compile-verified
//
#include <hip/hip_runtime.h>
#include <hip/hip_bf16.h>
#include <math.h>
#include <stdint.h>

// SAPS on MI455X, sort-free formulation:
//   ordered = [p_max, 0.2, 0.2, ...]  =>  cumsum analytic, sizes<=6, mask = top-k membership.
// K1: per-row online softmax (max, sumexp) + stable top-6 via async global->LDS double buffering.
// K2: stream cumsum (p_max + 0.2*j) and zero mask, float4 stores.
// K3: scatter <=6 ones per row into mask.

#define TAU      0.95f
#define RANK_PEN 0.2f
#define B_ROWS   4096
#define C_COLS   10000
#define CHUNK_F  1024                       // floats per async chunk (256 lanes x 4)
#define NCHUNK   ((C_COLS + CHUNK_F - 1) / CHUNK_F)   // 10
#define TOPK     6

__device__ __forceinline__ bool rank_before(float va, int ia, float vb, int ib) {
    // descending value, stable (smaller index first) -- matches jnp.argsort(-logits)
    return (va > vb) || ((va == vb) && (ia < ib));
}

__global__ __launch_bounds__(256) void saps_rowstats_kernel(
    const float* __restrict__ logits, const float* __restrict__ u,
    float* __restrict__ out_sizes, float* __restrict__ ws_pmax,
    int* __restrict__ ws_sz, int* __restrict__ ws_top)
{
    __shared__ __align__(16) float abuf[2][CHUNK_F];   // async double buffer (8 KB)
    __shared__ float sv[256 * TOPK];
    __shared__ int   si[256 * TOPK];
    __shared__ float sm[256];
    __shared__ float ss[256];

    const int tid = threadIdx.x;
    const int row = blockIdx.x;
    const uint64_t rowbase = (uint64_t)(uintptr_t)logits + (uint64_t)row * ((uint64_t)C_COLS * 4u);
    const uint32_t rowbytes = C_COLS * 4u;   // 40000, 16B-aligned rows

    // ---- issue chunk 0 (clamped tail addresses => every wave issues exactly one async op) ----
    {
        uint32_t boff = (uint32_t)(tid * 16);
        if (boff > rowbytes - 16u) boff = rowbytes - 16u;
        uint64_t g = rowbase + boff;
        uint32_t l = (uint32_t)(uintptr_t)&abuf[0][tid * 4];   // flat LDS addr low 32b == LDS offset
        asm volatile("global_load_async_to_lds_b128 %0, %1, off" :: "v"(l), "v"(g) : "memory");
    }

    float m = -INFINITY, s = 0.0f;
    float tv[TOPK]; int ti[TOPK];
#pragma unroll
    for (int j = 0; j < TOPK; ++j) { tv[j] = -INFINITY; ti[j] = 0x7fffffff; }

    for (int c = 0; c < NCHUNK; ++c) {
        if (c + 1 < NCHUNK) {
            uint32_t boff = (uint32_t)((c + 1) * CHUNK_F * 4 + tid * 16);
            if (boff > rowbytes - 16u) boff = rowbytes - 16u;
            uint64_t g = rowbase + boff;
            uint32_t l = (uint32_t)(uintptr_t)&abuf[(c + 1) & 1][tid * 4];
            asm volatile("global_load_async_to_lds_b128 %0, %1, off" :: "v"(l), "v"(g) : "memory");
            // two in flight; async loads complete in order => <=1 outstanding means chunk c is in LDS
            asm volatile("s_wait_asynccnt 0x1" ::: "memory");
        } else {
            asm volatile("s_wait_asynccnt 0x0" ::: "memory");
        }

        const int base = c * CHUNK_F + tid * 4;       // C_COLS % 4 == 0 => no ragged lane
        if (base < C_COLS) {
            const float4 v4 = *(const float4*)&abuf[c & 1][tid * 4];
            float vals[4] = { v4.x, v4.y, v4.z, v4.w };
#pragma unroll
            for (int e = 0; e < 4; ++e) {
                float v = vals[e]; int idx = base + e;
                // online softmax accumulation (p_max = 1 / sum(exp(x - max)))
                if (v > m) { s = s * __expf(m - v) + 1.0f; m = v; }
                else       { s += __expf(v - m); }
                // stable top-6 insert (indices per thread are strictly increasing)
                if (rank_before(v, idx, tv[TOPK - 1], ti[TOPK - 1])) {
                    tv[TOPK - 1] = v; ti[TOPK - 1] = idx;
#pragma unroll
                    for (int p = TOPK - 1; p > 0; --p) {
                        if (rank_before(tv[p], ti[p], tv[p - 1], ti[p - 1])) {
                            float fv = tv[p]; tv[p] = tv[p - 1]; tv[p - 1] = fv;
                            int  fi = ti[p]; ti[p] = ti[p - 1]; ti[p - 1] = fi;
                        }
                    }
                }
            }
        }
    }

    // ---- block reduction: (m, s) log-sum-exp combine + top-6 sorted-list merge tree ----
    sm[tid] = m; ss[tid] = s;
#pragma unroll
    for (int j = 0; j < TOPK; ++j) { sv[tid * TOPK + j] = tv[j]; si[tid * TOPK + j] = ti[j]; }
    __syncthreads();

    for (int stride = 128; stride >= 1; stride >>= 1) {
        if (tid < stride) {
            float ma = sm[tid], mb = sm[tid + stride];
            float M = fmaxf(ma, mb);
            ss[tid] = ss[tid] * __expf(ma - M) + ss[tid + stride] * __expf(mb - M);
            sm[tid] = M;

            const int a = tid * TOPK, b = (tid + stride) * TOPK;
            float rv[TOPK]; int ri[TOPK];
            int pa = 0, pb = 0;
#pragma unroll
            for (int o = 0; o < TOPK; ++o) {        // pa,pb <= 5 when read (pa+pb==o)
                float va = sv[a + pa]; int ia = si[a + pa];
                float vb = sv[b + pb]; int ib = si[b + pb];
                if (rank_before(va, ia, vb, ib)) { rv[o] = va; ri[o] = ia; ++pa; }
                else                             { rv[o] = vb; ri[o] = ib; ++pb; }
            }
#pragma unroll
            for (int o = 0; o < TOPK; ++o) { sv[a + o] = rv[o]; si[a + o] = ri[o]; }
        }
        __syncthreads();
    }

    if (tid == 0) {
        const float p = 1.0f / ss[0];              // p_max = softmax prob of argmax class
        int count;
        if (p > TAU) { count = 0; }
        else {
            count = 1;                              // cumsum[0] = p_max <= tau
            float ck = p;
            while (count < C_COLS) {                // <=5 iters; sequential adds match jnp.cumsum
                float c2 = ck + RANK_PEN;
                if (c2 <= TAU) { ck = c2; ++count; } else break;
            }
        }
        int sizes_base = count + 1; if (sizes_base > C_COLS) sizes_base = C_COLS;
        const int id = sizes_base - 1;
        float csum = p;
        for (int j = 1; j <= id; ++j) csum += RANK_PEN;          // cumsum[id]
        const float ok = (id == 0) ? p : RANK_PEN;               // ordered[id]
        const float V = (TAU - (csum - ok)) / ok;
        const int sz = sizes_base - ((u[row] >= V) ? 1 : 0);

        ws_pmax[row] = p;
        ws_sz[row]   = sz;
        out_sizes[row] = (float)sz;
#pragma unroll
        for (int j = 0; j < TOPK; ++j) ws_top[row * TOPK + j] = si[j];
    }
}

__global__ __launch_bounds__(256) void saps_write_kernel(
    const float* __restrict__ ws_pmax, float* __restrict__ cumsum, float* __restrict__ mask)
{
    const int row = blockIdx.y;
    const int q = blockIdx.x * blockDim.x + threadIdx.x;   // float4 index within row
    const int QPR = C_COLS / 4;                            // 2500
    __builtin_prefetch(ws_pmax + row, 0, 0);               // global_prefetch_b8
    if (q >= QPR) return;
    const float p = ws_pmax[row];
    const int j = q * 4;
    float4 v;
    v.x = p + RANK_PEN * (float)(j + 0);
    v.y = p + RANK_PEN * (float)(j + 1);
    v.z = p + RANK_PEN * (float)(j + 2);
    v.w = p + RANK_PEN * (float)(j + 3);
    const size_t off = (size_t)row * QPR + q;
    ((float4*)cumsum)[off] = v;                            // global_store_b128
    float4 z; z.x = 0.f; z.y = 0.f; z.z = 0.f; z.w = 0.f;
    ((float4*)mask)[off] = z;
}

__global__ __launch_bounds__(256) void saps_scatter_kernel(
    const int* __restrict__ ws_sz, const int* __restrict__ ws_top, float* __restrict__ mask)
{
    const int r = blockIdx.x * blockDim.x + threadIdx.x;
    if (r >= B_ROWS) return;
    int k = ws_sz[r]; k = k < 0 ? 0 : (k > TOPK ? TOPK : k);
    for (int j = 0; j < k; ++j) {
        const int cidx = ws_top[r * TOPK + j];
        mask[(size_t)r * C_COLS + cidx] = 1.0f;
    }
}

extern "C" void kernel_launch(void* const* d_in, const int* in_sizes, int n_in,
                              void* d_out, int out_size, void* d_ws, size_t ws_size,
                              hipStream_t stream)
{
    (void)in_sizes; (void)n_in; (void)out_size; (void)ws_size;
    const float* logits = (const float*)d_in[0];
    const float* u      = (const float*)d_in[1];

    float* out        = (float*)d_out;
    float* out_cumsum = out;                                       // B*C floats
    float* out_sizes  = out + (size_t)B_ROWS * C_COLS;             // B floats
    float* out_mask   = out + (size_t)B_ROWS * C_COLS + B_ROWS;    // B*C floats

    float* ws_pmax = (float*)d_ws;                                 // B
    int*   ws_sz   = (int*)(ws_pmax + B_ROWS);                     // B
    int*   ws_top  = ws_sz + B_ROWS;                               // B*6  (total 128 KB)

    saps_rowstats_kernel<<<dim3(B_ROWS), dim3(256), 0, stream>>>(
        logits, u, out_sizes, ws_pmax, ws_sz, ws_top);

    saps_write_kernel<<<dim3((C_COLS / 4 + 255) / 256, B_ROWS), dim3(256), 0, stream>>>(
        ws_pmax, out_cumsum, out_mask);

    saps_scatter_kernel<<<dim3((B_ROWS + 255) / 256), dim3(256), 0, stream>>>(
        ws_sz, ws_top, out_mask);
}